// WL1_64183991271884
// MI455X (gfx1250) — compile-verified
//
#include <hip/hip_runtime.h>
#include <stdint.h>

// Problem constants (fixed by the reference harness)
#define BATCH 8
#define CHANS 3
#define HH 512
#define WW 512
#define NPIX (HH*WW)            // 262144
#define CHW (CHANS*NPIX)        // 786432
#define KS 7
#define KPAD 3
#define KN 49

typedef __attribute__((ext_vector_type(2))) float v2f;
typedef __attribute__((ext_vector_type(8))) float v8f;

// ---------------------------------------------------------------------------
// Kernel 1: residuals + per-block sum / sumsq partials of res_sr
// grid (64, 8), block 256; each block handles 4096 contiguous pixels of batch b
// ---------------------------------------------------------------------------
__global__ void k_residuals(const float* __restrict__ sr,
                            const float* __restrict__ sre,
                            const float* __restrict__ hr,
                            float* __restrict__ res_sr,
                            float* __restrict__ res_ema,
                            float* __restrict__ bpart) {
  const int b = blockIdx.y;
  const int chunk = blockIdx.x;
  const int t = threadIdx.x;
  const size_t pb = (size_t)b * CHW;
  const size_t ob = (size_t)b * NPIX;
  const int base = chunk * 4096;
  const float inv255 = 1.0f / 255.0f;

  float s = 0.f, s2 = 0.f;
#pragma unroll 4
  for (int k = 0; k < 16; ++k) {
    const int p = base + t + k * 256;
    const float h0 = hr[pb + p];
    const float h1 = hr[pb + NPIX + p];
    const float h2 = hr[pb + 2 * NPIX + p];
    const float a0 = sr[pb + p];
    const float a1 = sr[pb + NPIX + p];
    const float a2 = sr[pb + 2 * NPIX + p];
    const float e0 = sre[pb + p];
    const float e1 = sre[pb + NPIX + p];
    const float e2 = sre[pb + 2 * NPIX + p];
    const float rs = (fabsf(h0 - a0) + fabsf(h1 - a1) + fabsf(h2 - a2)) * inv255;
    const float re = (fabsf(h0 - e0) + fabsf(h1 - e1) + fabsf(h2 - e2)) * inv255;
    res_sr[ob + p] = rs;
    res_ema[ob + p] = re;
    s += rs;
    s2 += rs * rs;
  }

  __shared__ float rbuf[512];
  rbuf[t] = s;
  rbuf[256 + t] = s2;
  __syncthreads();
  for (int o = 128; o > 0; o >>= 1) {
    if (t < o) {
      rbuf[t] += rbuf[t + o];
      rbuf[256 + t] += rbuf[256 + t + o];
    }
    __syncthreads();
  }
  if (t == 0) {
    bpart[2 * (b * 64 + chunk) + 0] = rbuf[0];
    bpart[2 * (b * 64 + chunk) + 1] = rbuf[256];
  }
}

// ---------------------------------------------------------------------------
// Kernel 2: per-batch patch_w = var(res_sr, ddof=1) ** 0.2
// ---------------------------------------------------------------------------
__global__ void k_patchw(const float* __restrict__ bpart,
                         float* __restrict__ patchw) {
  const int t = threadIdx.x;
  if (t < BATCH) {
    float s = 0.f, s2 = 0.f;
    for (int i = 0; i < 64; ++i) {
      s += bpart[2 * (t * 64 + i) + 0];
      s2 += bpart[2 * (t * 64 + i) + 1];
    }
    const float Nf = (float)NPIX;
    const float var = (s2 - s * s / Nf) / (Nf - 1.0f);
    patchw[t] = powf(var, 0.2f);
  }
}

// ---------------------------------------------------------------------------
// Kernel 3: 7x7 local variance via banded-matrix WMMA, masked weighted sum.
// grid (64, 8): blockIdx.x = 64x64 tile index, blockIdx.y = batch.
// Block = 256 threads = 8 waves; each wave does two 16x16 output subtiles.
// Box sum S = VB(16x22) * P(22x22) * HB(22x16), each banded 0/1, via
// V_WMMA_F32_16X16X4_F32 with K padded to 24.
// ---------------------------------------------------------------------------
__global__ void k_main(const float* __restrict__ res_sr,
                       const float* __restrict__ res_ema,
                       const float* __restrict__ patchw,
                       float* __restrict__ losspart) {
  __shared__ float tileP[72 * 72];         // reflected res_sr halo tile
  __shared__ float m1buf[8][2][24 * 16];   // per-wave intermediates (s1, s2)
  __shared__ float red[256];

  const int b = blockIdx.y;
  const int y0 = (blockIdx.x >> 3) * 64;
  const int x0 = (blockIdx.x & 7) * 64;
  const int tid = threadIdx.x;
  const int wv = tid >> 5;
  const int lane = tid & 31;
  const int lhalf = lane >> 4;
  const int nn = lane & 15;

  const float pw = patchw[b];
  const float* plane = res_sr + (size_t)b * NPIX;

  // ---- async gather of 72x72 reflected tile straight into LDS ----
  for (int idx = tid; idx < 72 * 72; idx += 256) {
    const int ly = idx / 72;
    const int lx = idx - ly * 72;
    int gy = y0 - KPAD + ly;
    int gx = x0 - KPAD + lx;
    gy = (gy < 0) ? -gy : ((gy >= HH) ? (2 * HH - 2 - gy) : gy);
    gx = (gx < 0) ? -gx : ((gx >= WW) ? (2 * WW - 2 - gx) : gx);
    const uint32_t voff = (uint32_t)(gy * WW + gx) * 4u;
    const uint32_t ldso = (uint32_t)(uintptr_t)&tileP[idx];
    asm volatile("global_load_async_to_lds_b32 %0, %1, %2"
                 :
                 : "v"(ldso), "v"(voff), "s"(plane)
                 : "memory");
  }
  asm volatile("s_wait_asynccnt 0" ::: "memory");
  __syncthreads();

  // ---- banded 0/1 constant matrices, one array serves both HB (B-op) and
  //      VB (A-op): entry(g,half) covers index j = kc*4 + g + 2*half,
  //      value = (nn <= j <= nn+6) ----
  v2f bandc[6];
#pragma unroll
  for (int kc = 0; kc < 6; ++kc) {
    v2f h;
#pragma unroll
    for (int g = 0; g < 2; ++g) {
      const int j = kc * 4 + g + 2 * lhalf;
      h[g] = (j >= nn && j <= nn + 6) ? 1.0f : 0.0f;
    }
    bandc[kc] = h;
  }

  // zero intermediate rows 22..23 once (K padded to 24; avoid 0*NaN in WMMA)
  m1buf[wv][0][352 + lane] = 0.0f;
  m1buf[wv][1][352 + lane] = 0.0f;

  float lsum = 0.0f;

#pragma unroll
  for (int sti = 0; sti < 2; ++sti) {
    const int st = wv + sti * 8;          // subtile 0..15 within 64x64 block
    const int sy = (st >> 2) * 16;
    const int sx = (st & 3) * 16;

    // ---- pass 1: M1 = P x HB (horizontal 7-sum), rows 0..21 via two tiles ----
#pragma unroll
    for (int rbi = 0; rbi < 2; ++rbi) {
      const int rb = rbi * 6;             // row base 0 or 6
      const int pr = sy + rb + nn;        // A row = P row in 72-tile
      v8f acc1 = {0.f, 0.f, 0.f, 0.f, 0.f, 0.f, 0.f, 0.f};
      v8f acc2 = {0.f, 0.f, 0.f, 0.f, 0.f, 0.f, 0.f, 0.f};
#pragma unroll
      for (int kc = 0; kc < 6; ++kc) {
        const int pc = sx + kc * 4 + 2 * lhalf;
        v2f a;
        a[0] = tileP[pr * 72 + pc];
        a[1] = tileP[pr * 72 + pc + 1];
        const v2f asq = a * a;
        acc1 = __builtin_amdgcn_wmma_f32_16x16x4_f32(
            false, a, false, bandc[kc], (short)0, acc1, false, false);
        acc2 = __builtin_amdgcn_wmma_f32_16x16x4_f32(
            false, asq, false, bandc[kc], (short)0, acc2, false, false);
      }
#pragma unroll
      for (int g = 0; g < 8; ++g) {
        const int row = rb + g + 8 * lhalf;
        m1buf[wv][0][row * 16 + nn] = acc1[g];
        m1buf[wv][1][row * 16 + nn] = acc2[g];
      }
    }

    // ---- pass 2: S = VB x M1 (vertical 7-sum) ----
    v8f S1 = {0.f, 0.f, 0.f, 0.f, 0.f, 0.f, 0.f, 0.f};
    v8f S2 = {0.f, 0.f, 0.f, 0.f, 0.f, 0.f, 0.f, 0.f};
#pragma unroll
    for (int kc = 0; kc < 6; ++kc) {
      const int r0 = kc * 4 + 2 * lhalf;
      v2f b1, b2;
      b1[0] = m1buf[wv][0][(r0 + 0) * 16 + nn];
      b1[1] = m1buf[wv][0][(r0 + 1) * 16 + nn];
      b2[0] = m1buf[wv][1][(r0 + 0) * 16 + nn];
      b2[1] = m1buf[wv][1][(r0 + 1) * 16 + nn];
      S1 = __builtin_amdgcn_wmma_f32_16x16x4_f32(
          false, bandc[kc], false, b1, (short)0, S1, false, false);
      S2 = __builtin_amdgcn_wmma_f32_16x16x4_f32(
          false, bandc[kc], false, b2, (short)0, S2, false, false);
    }

    // ---- epilogue: pix_w, mask, weighted accumulation ----
#pragma unroll
    for (int g = 0; g < 8; ++g) {
      const int m = g + 8 * lhalf;
      const float s1 = S1[g];
      const float s2 = S2[g];
      const float pixw = (s2 - s1 * s1 * (1.0f / (float)KN)) * (1.0f / (float)(KN - 1));
      const float rs = tileP[(sy + KPAD + m) * 72 + (sx + KPAD + nn)];
      const int oy = y0 + sy + m;
      const int ox = x0 + sx + nn;
      const float re = res_ema[(size_t)b * NPIX + oy * WW + ox];
      const float wgt = (rs < re) ? 0.0f : (pw * pixw);
      lsum += wgt * rs;
    }
  }

  // ---- deterministic block reduction ----
  red[tid] = lsum;
  __syncthreads();
  for (int o = 128; o > 0; o >>= 1) {
    if (tid < o) red[tid] += red[tid + o];
    __syncthreads();
  }
  if (tid == 0) losspart[blockIdx.y * 64 + blockIdx.x] = red[0];
}

// ---------------------------------------------------------------------------
// Kernel 4: final deterministic reduction of 512 partials -> scalar loss
// ---------------------------------------------------------------------------
__global__ void k_final(const float* __restrict__ losspart,
                        float* __restrict__ out) {
  __shared__ float red[256];
  const int t = threadIdx.x;
  red[t] = losspart[t] + losspart[t + 256];
  __syncthreads();
  for (int o = 128; o > 0; o >>= 1) {
    if (t < o) red[t] += red[t + o];
    __syncthreads();
  }
  if (t == 0) {
    // mean over B*C*H*W of w*|sr-hr|, with sum_c|sr-hr| = 255*res_sr
    out[0] = red[0] * (255.0f / (float)((size_t)BATCH * CHW));
  }
}

// ---------------------------------------------------------------------------
extern "C" void kernel_launch(void* const* d_in, const int* in_sizes, int n_in,
                              void* d_out, int out_size, void* d_ws, size_t ws_size,
                              hipStream_t stream) {
  const float* sr = (const float*)d_in[0];
  const float* sre = (const float*)d_in[1];
  const float* hr = (const float*)d_in[2];
  // d_in[3] = ksize (always 7 for this harness) — baked into the kernels.
  (void)in_sizes; (void)n_in; (void)out_size; (void)ws_size;

  float* ws = (float*)d_ws;
  float* res_sr = ws;                         // 2,097,152 floats
  float* res_ema = ws + 2097152;              // 2,097,152 floats
  float* bpart = ws + 4194304;                // 1024 floats (512 blocks x {s,s2})
  float* patchw = ws + 4195328;               // 8 floats
  float* losspart = ws + 4195336;             // 512 floats
  float* out = (float*)d_out;

  k_residuals<<<dim3(64, 8), 256, 0, stream>>>(sr, sre, hr, res_sr, res_ema, bpart);
  k_patchw<<<1, 64, 0, stream>>>(bpart, patchw);
  k_main<<<dim3(64, 8), 256, 0, stream>>>(res_sr, res_ema, patchw, losspart);
  k_final<<<1, 256, 0, stream>>>(losspart, out);
}